// ContextQueryAttention_30734785970908
// MI455X (gfx1250) — compile-verified
//
#include <hip/hip_runtime.h>
#include <hip/hip_bf16.h>
#include <math.h>

// Context-Query attention (BiDAF/QANet style) for MI455X (gfx1250, wave32).
// B=16, Lc=2048, Lq=512, H=128, fp32 throughout using V_WMMA_F32_16X16X4_F32.

typedef __attribute__((ext_vector_type(2))) float v2f;
typedef __attribute__((ext_vector_type(8))) float v8f;

#define B_  16
#define LC_ 2048
#define LQ_ 512
#define H_  128

static __device__ __forceinline__ v8f wmma4_f32(v2f a, v2f b, v8f c) {
  // 8 args: (neg_a, A, neg_b, B, c_mod, C, reuse_a, reuse_b)
  return __builtin_amdgcn_wmma_f32_16x16x4_f32(false, a, false, b, (short)0, c,
                                               false, false);
}

static __device__ __forceinline__ float clip15(float x) {
  return fminf(fmaxf(x, -15.0f), 15.0f);
}

// ---------------------------------------------------------------------------
// Kernel 1: rC[b,c] = ctx[b,c,:]·w_C ; rQ[b,q] = query[b,q,:]·w_Q
// ---------------------------------------------------------------------------
__global__ void k_rowdots(const float* __restrict__ ctx,
                          const float* __restrict__ query,
                          const float* __restrict__ w_C,
                          const float* __restrict__ w_Q,
                          float* __restrict__ rC, float* __restrict__ rQ) {
  int idx = blockIdx.x * blockDim.x + threadIdx.x;
  const int nC = B_ * LC_;
  const int nQ = B_ * LQ_;
  if (idx < nC) {
    const float* row = ctx + (size_t)idx * H_;
    float acc = 0.f;
#pragma unroll 8
    for (int h = 0; h < H_; ++h) acc += row[h] * w_C[h];
    rC[idx] = acc;
  } else if (idx < nC + nQ) {
    int j = idx - nC;
    const float* row = query + (size_t)j * H_;
    float acc = 0.f;
#pragma unroll 8
    for (int h = 0; h < H_; ++h) acc += row[h] * w_Q[h];
    rQ[j] = acc;
  }
}

// ---------------------------------------------------------------------------
// Kernel 2: S[b,c,q] = clip( (w_CQ*ctx[b,c,:])·query[b,q,:] + rC + rQ + bias )
// One block per (b, 16-row ctx tile). ctx tile * w_CQ staged in LDS, shared by
// the 8 waves; wave w covers q in [64w, 64w+64) with 4 fp32 WMMA accumulators.
// ---------------------------------------------------------------------------
__global__ void k_similarity(const float* __restrict__ ctx,
                             const float* __restrict__ query,
                             const float* __restrict__ w_CQ,
                             const float* __restrict__ rC,
                             const float* __restrict__ rQ,
                             const float* __restrict__ bias,
                             float* __restrict__ S) {
  __shared__ float As[16][132];  // 16 rows x 128 (+4 pad), w_CQ pre-applied

  const int b  = blockIdx.x / (LC_ / 16);
  const int ct = blockIdx.x % (LC_ / 16);
  const int c0 = ct * 16;
  const int tid = threadIdx.x;

  // cooperative stage of the scaled ctx tile
  for (int i = tid; i < 16 * H_; i += 256) {
    int r = i >> 7, h = i & (H_ - 1);
    As[r][h] = ctx[((size_t)b * LC_ + c0 + r) * H_ + h] * w_CQ[h];
  }
  __syncthreads();

  const int wave  = tid >> 5;
  const int lane  = tid & 31;
  const int lm    = lane & 15;
  const int lh    = lane >> 4;       // 0: K pair {0,1}; 1: K pair {2,3}
  const int q0w   = wave * 64;

  v8f acc0 = {}, acc1 = {}, acc2 = {}, acc3 = {};

  const float* qbase = query + ((size_t)b * LQ_ + q0w + lm) * H_ + 2 * lh;

#pragma unroll 4
  for (int hs = 0; hs < H_; hs += 4) {
    v2f a = *(const v2f*)(&As[lm][hs + 2 * lh]);
    v2f b0 = *(const v2f*)(qbase + (size_t)(0 * 16) * H_ + hs);
    v2f b1 = *(const v2f*)(qbase + (size_t)(1 * 16) * H_ + hs);
    v2f b2 = *(const v2f*)(qbase + (size_t)(2 * 16) * H_ + hs);
    v2f b3 = *(const v2f*)(qbase + (size_t)(3 * 16) * H_ + hs);
    acc0 = wmma4_f32(a, b0, acc0);
    acc1 = wmma4_f32(a, b1, acc1);
    acc2 = wmma4_f32(a, b2, acc2);
    acc3 = wmma4_f32(a, b3, acc3);
  }

  const float bv = bias[0];
  v8f accs[4] = {acc0, acc1, acc2, acc3};
#pragma unroll
  for (int t = 0; t < 4; ++t) {
    const int q = q0w + t * 16 + lm;
    const float rqv = rQ[b * LQ_ + q] + bv;
#pragma unroll
    for (int r = 0; r < 8; ++r) {
      const int c = c0 + r + 8 * lh;
      float v = accs[t][r] + rC[b * LC_ + c] + rqv;
      S[((size_t)b * LC_ + c) * LQ_ + q] = clip15(v);
    }
  }
}

// ---------------------------------------------------------------------------
// Kernel 3: column softmax stats (axis=1, over ctx, mask = ctx_mask[b,c]).
// Thread per (b,q): online max/sum over c. Coalesced in q.
// ---------------------------------------------------------------------------
__global__ void k_colstats(const float* __restrict__ S,
                           const float* __restrict__ ctx_mask,
                           float* __restrict__ colmax,
                           float* __restrict__ coldenom) {
  int idx = blockIdx.x * blockDim.x + threadIdx.x;
  if (idx >= B_ * LQ_) return;
  const int b = idx / LQ_;
  const int q = idx % LQ_;
  const float* Sp = S + (size_t)b * LC_ * LQ_ + q;
  const float* cm = ctx_mask + b * LC_;
  float m = -1e30f, s = 0.f;
  for (int c = 0; c < LC_; ++c) {
    const float mk = cm[c];
    const float x = Sp[(size_t)c * LQ_] * mk;
    const float nm = fmaxf(m, x);
    s = s * __expf(m - nm) + mk * __expf(x - nm);
    m = nm;
  }
  colmax[idx] = m;
  coldenom[idx] = s + 1e-6f;
}

// ---------------------------------------------------------------------------
// Kernel 4: row softmax stats (axis=2, over query, mask = query_mask[b,q]).
// One wave per (b,c) row, wave32 shfl_xor reduction.
// ---------------------------------------------------------------------------
__global__ void k_rowstats(const float* __restrict__ S,
                           const float* __restrict__ query_mask,
                           float* __restrict__ rowmax,
                           float* __restrict__ rowdenom) {
  const int waveId = (blockIdx.x * blockDim.x + threadIdx.x) >> 5;
  const int lane = threadIdx.x & 31;
  if (waveId >= B_ * LC_) return;
  const int b = waveId / LC_;
  const int c = waveId % LC_;
  const float* Sp = S + ((size_t)b * LC_ + c) * LQ_;
  const float* qm = query_mask + b * LQ_;

  float m = -1e30f, s = 0.f;
  for (int q = lane; q < LQ_; q += 32) {
    const float mk = qm[q];
    const float x = Sp[q] * mk;
    const float nm = fmaxf(m, x);
    s = s * __expf(m - nm) + mk * __expf(x - nm);
    m = nm;
  }
#pragma unroll
  for (int off = 16; off > 0; off >>= 1) {
    const float om = __shfl_xor(m, off, 32);
    const float os = __shfl_xor(s, off, 32);
    const float nm = fmaxf(m, om);
    s = s * __expf(m - nm) + os * __expf(om - nm);
    m = nm;
  }
  if (lane == 0) {
    rowmax[waveId] = m;
    rowdenom[waveId] = s + 1e-6f;
  }
}

// ---------------------------------------------------------------------------
// Kernel 5: A[b,q,h] = sum_c s_ctx[b,c,q] * ctx[b,c,h]  (= s_ctx^T @ ctx)
// One wave per (b, 16-q tile): 8 n-tiles of H, K loop over c in steps of 4.
// s_ctx numerators built in the A-fragment; 1/denom applied in epilogue.
// ---------------------------------------------------------------------------
__global__ void k_ctxpool(const float* __restrict__ S,
                          const float* __restrict__ ctx,
                          const float* __restrict__ ctx_mask,
                          const float* __restrict__ colmax,
                          const float* __restrict__ coldenom,
                          float* __restrict__ Abuf) {
  const int waveId = (blockIdx.x * blockDim.x + threadIdx.x) >> 5;
  const int lane = threadIdx.x & 31;
  const int lm = lane & 15;
  const int lh = lane >> 4;
  if (waveId >= B_ * (LQ_ / 16)) return;
  const int b = waveId / (LQ_ / 16);
  const int q0 = (waveId % (LQ_ / 16)) * 16;

  const float cmax = colmax[b * LQ_ + q0 + lm];  // per A-fragment row (M=lm)

  v8f acc[8];
#pragma unroll
  for (int t = 0; t < 8; ++t) acc[t] = (v8f){};

  for (int c0 = 0; c0 < LC_; c0 += 4) {
    const int ck = c0 + 2 * lh;  // this lane's K pair: ck, ck+1
    v2f cmv = *(const v2f*)(ctx_mask + b * LC_ + ck);
    const float* Sp = S + ((size_t)b * LC_ + ck) * LQ_ + q0 + lm;
    const float s0 = Sp[0];
    const float s1 = Sp[LQ_];
    v2f a;
    a.x = __expf(s0 * cmv.x - cmax) * cmv.x;
    a.y = __expf(s1 * cmv.y - cmax) * cmv.y;

    const float* Cp = ctx + ((size_t)b * LC_ + ck) * H_ + lm;
#pragma unroll
    for (int t = 0; t < 8; ++t) {
      v2f bb;
      bb.x = Cp[t * 16];
      bb.y = Cp[t * 16 + H_];
      acc[t] = wmma4_f32(a, bb, acc[t]);
    }
  }

#pragma unroll
  for (int r = 0; r < 8; ++r) {
    const int q = q0 + r + 8 * lh;
    const float inv = 1.0f / coldenom[b * LQ_ + q];
#pragma unroll
    for (int t = 0; t < 8; ++t)
      Abuf[((size_t)b * LQ_ + q) * H_ + t * 16 + lm] = acc[t][r] * inv;
  }
}

// ---------------------------------------------------------------------------
// Kernel 6: per 16-row ctx tile:
//   P  = s_query @ query          (16 x 128)
//   Qc = s_query @ Abuf           (16 x 128)
//   out[b,c,:] = [ctx, P, ctx*P, ctx*Qc]
// s_query numerators built in the A-fragment; 1/rowdenom in epilogue.
// ---------------------------------------------------------------------------
__global__ void k_output(const float* __restrict__ S,
                         const float* __restrict__ ctx,
                         const float* __restrict__ query,
                         const float* __restrict__ Abuf,
                         const float* __restrict__ query_mask,
                         const float* __restrict__ rowmax,
                         const float* __restrict__ rowdenom,
                         float* __restrict__ out) {
  const int waveId = (blockIdx.x * blockDim.x + threadIdx.x) >> 5;
  const int lane = threadIdx.x & 31;
  const int lm = lane & 15;
  const int lh = lane >> 4;
  if (waveId >= B_ * (LC_ / 16)) return;
  const int b = waveId / (LC_ / 16);
  const int c0 = (waveId % (LC_ / 16)) * 16;

  const float rmax = rowmax[b * LC_ + c0 + lm];  // per A-fragment row (M=lm)

  v8f accP[8], accQ[8];
#pragma unroll
  for (int t = 0; t < 8; ++t) { accP[t] = (v8f){}; accQ[t] = (v8f){}; }

  const float* Srow = S + ((size_t)b * LC_ + c0 + lm) * LQ_ + 2 * lh;

  for (int q = 0; q < LQ_; q += 4) {
    v2f qmv = *(const v2f*)(query_mask + b * LQ_ + q + 2 * lh);
    v2f sv  = *(const v2f*)(Srow + q);
    v2f a;
    a.x = __expf(sv.x * qmv.x - rmax) * qmv.x;
    a.y = __expf(sv.y * qmv.y - rmax) * qmv.y;

    const size_t krow = (size_t)b * LQ_ + q + 2 * lh;
    const float* Qp = query + krow * H_ + lm;
    const float* Ap = Abuf + krow * H_ + lm;
#pragma unroll
    for (int t = 0; t < 8; ++t) {
      v2f bp, bq;
      bp.x = Qp[t * 16];
      bp.y = Qp[t * 16 + H_];
      bq.x = Ap[t * 16];
      bq.y = Ap[t * 16 + H_];
      accP[t] = wmma4_f32(a, bp, accP[t]);
      accQ[t] = wmma4_f32(a, bq, accQ[t]);
    }
  }

#pragma unroll
  for (int r = 0; r < 8; ++r) {
    const int c = c0 + r + 8 * lh;
    const float inv = 1.0f / rowdenom[b * LC_ + c];
    float* ob = out + ((size_t)b * LC_ + c) * (4 * H_);
    const float* cb = ctx + ((size_t)b * LC_ + c) * H_;
#pragma unroll
    for (int t = 0; t < 8; ++t) {
      const int h = t * 16 + lm;
      const float cv = cb[h];
      const float pv = accP[t][r] * inv;
      const float qv = accQ[t][r] * inv;
      ob[h] = cv;
      ob[H_ + h] = pv;
      ob[2 * H_ + h] = cv * pv;
      ob[3 * H_ + h] = cv * qv;
    }
  }
}

// ---------------------------------------------------------------------------
extern "C" void kernel_launch(void* const* d_in, const int* in_sizes, int n_in,
                              void* d_out, int out_size, void* d_ws,
                              size_t ws_size, hipStream_t stream) {
  const float* ctx        = (const float*)d_in[0];
  const float* query      = (const float*)d_in[1];
  const float* ctx_mask   = (const float*)d_in[2];
  const float* query_mask = (const float*)d_in[3];
  const float* w_C        = (const float*)d_in[4];
  const float* w_Q        = (const float*)d_in[5];
  const float* w_CQ       = (const float*)d_in[6];
  const float* bias       = (const float*)d_in[7];
  float* out = (float*)d_out;

  // workspace layout (floats)
  float* ws = (float*)d_ws;
  float* Sws      = ws;                             // B*Lc*Lq   (64 MiB)
  float* rC       = Sws + (size_t)B_ * LC_ * LQ_;   // B*Lc
  float* rQ       = rC + B_ * LC_;                  // B*Lq
  float* colmax   = rQ + B_ * LQ_;                  // B*Lq
  float* coldenom = colmax + B_ * LQ_;              // B*Lq
  float* rowmax   = coldenom + B_ * LQ_;            // B*Lc
  float* rowdenom = rowmax + B_ * LC_;              // B*Lc
  float* Abuf     = rowdenom + B_ * LC_;            // B*Lq*H    (4 MiB)

  (void)in_sizes; (void)n_in; (void)out_size; (void)ws_size;

  // 1) row dots
  {
    int n = B_ * LC_ + B_ * LQ_;
    k_rowdots<<<(n + 255) / 256, 256, 0, stream>>>(ctx, query, w_C, w_Q, rC, rQ);
  }
  // 2) similarity S (clipped), WMMA + LDS staging
  k_similarity<<<B_ * (LC_ / 16), 256, 0, stream>>>(ctx, query, w_CQ, rC, rQ,
                                                    bias, Sws);
  // 3) column softmax stats
  k_colstats<<<(B_ * LQ_ + 255) / 256, 256, 0, stream>>>(Sws, ctx_mask, colmax,
                                                         coldenom);
  // 4) row softmax stats
  k_rowstats<<<(B_ * LC_) / 8, 256, 0, stream>>>(Sws, query_mask, rowmax,
                                                 rowdenom);
  // 5) A = s_ctx^T @ ctx, WMMA
  k_ctxpool<<<(B_ * (LQ_ / 16) + 7) / 8, 256, 0, stream>>>(
      Sws, ctx, ctx_mask, colmax, coldenom, Abuf);
  // 6) P, Qc and final concat, WMMA
  k_output<<<(B_ * (LC_ / 16)) / 8, 256, 0, stream>>>(
      Sws, ctx, query, Abuf, query_mask, rowmax, rowdenom, out);
}